// Poisson_79697413145338
// MI455X (gfx1250) — compile-verified
//
#include <hip/hip_runtime.h>
#include <cstdint>

// Poisson spike encoder for (16,1,224,224) f32 -> (100, 16,1,224,224) f32 {0,1}.
// Store-bandwidth bound: 321 MB out @ 23.3 TB/s ~= 13.8 us floor.
// CDNA5 path: per-thread spike bitmask -> transposed LDS tile -> wave32 issues
// GLOBAL_STORE_ASYNC_FROM_LDS_B128 (one 512B output row per instruction,
// tracked on ASYNCcnt) so the write stream drains while other waves run the
// VALU/trans-heavy RNG loop. WMMA is inapplicable (no matrix op in this ref).

#define T_WIN 100
#define PIX_PER_BLOCK 128          // 4 x wave32
#define WAVES_PER_BLOCK 4
#define ROWS_PER_WAVE (T_WIN / WAVES_PER_BLOCK)   // 25 output rows per wave

__device__ __forceinline__ uint32_t mix32(uint32_t x) {
  x ^= x >> 16; x *= 0x7feb352dU;
  x ^= x >> 15; x *= 0x846ca68bU;
  x ^= x >> 16; return x;
}

// Counter-based uniform in (0,1): deterministic, no cross-call state.
__device__ __forceinline__ float rng_uniform(uint32_t& s) {
  s = s * 747796405u + 2891336453u;      // LCG advance
  uint32_t h = mix32(s);                  // output mix
  return (float)(h >> 8) * 5.9604644775390625e-08f   // * 2^-24
         + 2.9802322387695312e-08f;                   // keep > 0 for log()
}

// Poisson(lam) sample.
//  lam < 16 : exponential-gap (Knuth in log domain) -- count unit-rate
//             arrivals in [0, lam); mean cost ~ lam+1 log-evals.
//  lam >= 16: normal approximation
__device__ __forceinline__ int poisson_sample(float lam, uint32_t& s) {
  if (lam < 16.0f) {
    int k = -1;
    float acc = 0.0f;
    do {
      ++k;
      acc -= __logf(rng_uniform(s));
    } while (acc < lam && k < 256);
    return k;
  }
  float z = (rng_uniform(s) + rng_uniform(s) + rng_uniform(s) - 1.5f) * 2.0f;
  float k = lam + z * __fsqrt_rn(lam);
  return (k < 1.0f) ? 1 : (int)(k + 0.5f);
}

__global__ __launch_bounds__(PIX_PER_BLOCK)
void poisson_encode_kernel(const float* __restrict__ img,
                           float* __restrict__ out, int N) {
  // Transposed staging tile: row t holds 128 consecutive output pixels.
  __shared__ float tile[T_WIN * PIX_PER_BLOCK];   // 51.2 KB of 320 KB WGP LDS

  const int tid  = threadIdx.x;
  const int base = blockIdx.x * PIX_PER_BLOCK;
  const int i    = base + tid;

  float x = 0.0f;
  if (i < N) x = img[i];                 // 3.2 MB total: L2-resident

  // --- generate spike bitmask (bit t => spike in output row t) ---
  uint32_t mask[4] = {0u, 0u, 0u, 0u};
  if (x > 0.0f) {
    // Raw v_rcp_f32: lambda feeds a stochastic sampler, the IEEE-exact
    // division expansion (v_div_scale chain) is wasted VALU here.
    float lam = __builtin_amdgcn_rcpf(x);   // RATIO == 1.0
    uint32_t s = mix32((uint32_t)i ^ 0xA511E9B3u);
    int c = 0;                           // cumulative spike time
    while (c < T_WIN) {
      int k = poisson_sample(lam, s);
      if (k < 1) k = 1;                  // reference clamps 0-intervals to 1
      c += k;
      if (c > T_WIN) break;              // times >= T+1 are discarded
      mask[(c - 1) >> 5] |= 1u << ((c - 1) & 31);
    }
  }

  // --- expand bitmask into transposed LDS tile ---
#pragma unroll
  for (int t = 0; t < T_WIN; ++t) {
    tile[t * PIX_PER_BLOCK + tid] =
        ((mask[t >> 5] >> (t & 31)) & 1u) ? 1.0f : 0.0f;
  }
  __syncthreads();

  // --- drain tile with async LDS->global stores (ASYNCcnt path) ---
  if (base + PIX_PER_BLOCK <= N) {
    const int wave = tid >> 5;
    const int lane = tid & 31;
#pragma unroll 5
    for (int r = 0; r < ROWS_PER_WAVE; ++r) {
      const int t = wave * ROWS_PER_WAVE + r;
      // 32 lanes x B128 = 512 B = one full contiguous output row slice.
      uint64_t gaddr = (uint64_t)(uintptr_t)
          (out + (size_t)t * (size_t)N + (size_t)(base + lane * 4));
      uint32_t laddr = (uint32_t)(uintptr_t)
          (&tile[t * PIX_PER_BLOCK + lane * 4]);   // low 32b of generic = LDS offset
      asm volatile("global_store_async_from_lds_b128 %0, %1, off"
                   :: "v"(gaddr), "v"(laddr)
                   : "memory");
    }
    // LDS must stay valid until async stores complete.
#if __has_builtin(__builtin_amdgcn_s_wait_asynccnt)
    __builtin_amdgcn_s_wait_asynccnt(0);
#else
    asm volatile("s_wait_asynccnt 0x0" ::: "memory");
#endif
  } else if (i < N) {
    // Tail block (not hit for N=802816, which is a multiple of 128).
    for (int t = 0; t < T_WIN; ++t)
      out[(size_t)t * (size_t)N + i] = tile[t * PIX_PER_BLOCK + tid];
  }
}

extern "C" void kernel_launch(void* const* d_in, const int* in_sizes, int n_in,
                              void* d_out, int out_size, void* d_ws, size_t ws_size,
                              hipStream_t stream) {
  const float* img = (const float*)d_in[0];
  float* out = (float*)d_out;
  const int N = in_sizes[0];             // 802816
  dim3 block(PIX_PER_BLOCK);
  dim3 grid((N + PIX_PER_BLOCK - 1) / PIX_PER_BLOCK);   // 6272 blocks
  hipLaunchKernelGGL(poisson_encode_kernel, grid, block, 0, stream,
                     img, out, N);
}